// lc_layer_2439541424305
// MI455X (gfx1250) — compile-verified
//
#include <hip/hip_runtime.h>
#include <cstdint>

// ---------------------------------------------------------------------------
// LC layer: outputx/outputy = per-pixel 2x2-ish complex transfer applied to
// (imagex, imagey), coefficients from a 28x28 voltage grid upsampled x20.
// Phase 1: 784-cell coefficient kernel (all fp64 transcendentals, -> d_ws).
// Phase 2: memory-bound fp64 complex stream, one block per image row.
//   - coefficients staged to LDS via CDNA5 global_load_async_to_lds_b128
//   - image traffic via b128 nontemporal loads/stores (read/write-once stream)
// ---------------------------------------------------------------------------

#define IMG    560
#define NCELL  28
#define SCALE  20
#define BATCH  16
#define PI_    3.14159265358979323846

typedef double __attribute__((ext_vector_type(2))) cplx;  // (re, im), 16B

// ------------------------- coefficient kernel ------------------------------
__global__ void lc_coef_kernel(const double* __restrict__ V,
                               double* __restrict__ tab) {
    int c = blockIdx.x * blockDim.x + threadIdx.x;
    if (c >= NCELL * NCELL) return;

    double w = V[c];
    double theta = 0.0;
    if (w > 1.0)                                   // (w > VC) mask, VC = VO = 1
        theta = PI_ * 0.5 - 2.0 * atan(exp(-(w - 1.0)));

    double ct = cos(theta) / 1.2;                  // cos(theta)/ne
    double st = sin(theta) / 1.06;                 // sin(theta)/no
    double ne_t = sqrt(1.0 / (ct * ct + st * st)); // effective index (real)

    const double pidl = PI_ * 1e-05 / 6.328e-07;   // pi*d/lambda
    double beta  = pidl * (ne_t - 1.06);
    double phi   = pidl * (ne_t + 1.06);
    double gamma = sqrt(PI_ * PI_ * 0.25 + beta * beta);

    double sg = sin(gamma), cg = cos(gamma);
    double A  = (PI_ / (2.0 * gamma)) * sg;        // half_pi_over_g * sin_g
    double bg = (beta / gamma) * sg;               // (beta/gamma) * sin_g
    double er = cos(phi), ei = -sin(phi);          // exp(-i*phi)

    double* t = tab + (size_t)c * 6;
    t[0] = er * A;                t[1] = ei * A;                 // P1 = E*A
    t[2] = er * cg - ei * bg;     t[3] = er * bg + ei * cg;      // P2 = E*( cg+i*bg)
    t[4] = -er * cg - ei * bg;    t[5] = er * bg - ei * cg;      // P3 = E*(-cg+i*bg)
}

// --------------------------- streaming kernel ------------------------------
// One block per (batch, row): blockIdx.x in [0, BATCH*IMG). 256 threads cover
// 560 columns (cols tid, tid+256, tid+512). The 28 cells of this row's
// coefficient stripe (28*48B = 1344B = 84 x 16B) are staged into LDS with the
// CDNA5 async copy path, then read via LDS broadcast.
__global__ __launch_bounds__(256) void lc_apply_kernel(
    const cplx* __restrict__ imx, const cplx* __restrict__ imy,
    const double* __restrict__ tab,
    cplx* __restrict__ outx, cplx* __restrict__ outy) {

    __shared__ __align__(16) double s_tab[NCELL * 6];   // 1344 bytes

    const int tid     = threadIdx.x;
    const int rowg    = blockIdx.x;        // b*IMG + r
    const int r       = rowg % IMG;
    const int cellrow = r / SCALE;         // which row of the 28x28 grid

    // --- async stage: 84 lanes x 16B = 1344B, gfx1250 TDM-class copy path ---
    if (tid < 84) {
        unsigned lds_off =
            (unsigned)(uintptr_t)(&s_tab[0]) + (unsigned)tid * 16u;
        unsigned long long gaddr =
            (unsigned long long)(uintptr_t)(tab + (size_t)cellrow * NCELL * 6)
            + (unsigned long long)tid * 16ull;
        asm volatile("global_load_async_to_lds_b128 %0, %1, off"
                     :: "v"(lds_off), "v"(gaddr)
                     : "memory");
    }
    asm volatile("s_wait_asynccnt 0" ::: "memory");   // this wave's LDS writes done
    __syncthreads();                                  // publish to all 8 waves

    const size_t rowbase = (size_t)rowg * IMG;

#pragma unroll
    for (int k = 0; k < 3; ++k) {
        int col = tid + k * 256;
        if (col < IMG) {
            const double* p = &s_tab[(col / SCALE) * 6];
            double P1r = p[0], P1i = p[1];
            double P2r = p[2], P2i = p[3];
            double P3r = p[4], P3i = p[5];

            size_t idx = rowbase + (size_t)col;
            cplx x = __builtin_nontemporal_load(&imx[idx]);  // b128, th:NT
            cplx y = __builtin_nontemporal_load(&imy[idx]);

            // outx = x*P1 + y*P2
            double oxr = x[0]*P1r - x[1]*P1i + y[0]*P2r - y[1]*P2i;
            double oxi = x[0]*P1i + x[1]*P1r + y[0]*P2i + y[1]*P2r;

            // outy = x * (P3 + y*P1)   (matches reference parenthesization)
            double tr = P3r + y[0]*P1r - y[1]*P1i;
            double ti = P3i + y[0]*P1i + y[1]*P1r;
            double oyr = x[0]*tr - x[1]*ti;
            double oyi = x[0]*ti + x[1]*tr;

            cplx ox; ox[0] = oxr; ox[1] = oxi;
            cplx oy; oy[0] = oyr; oy[1] = oyi;
            __builtin_nontemporal_store(ox, &outx[idx]);     // b128, th:NT
            __builtin_nontemporal_store(oy, &outy[idx]);
        }
    }
}

// ------------------------------- launcher ----------------------------------
extern "C" void kernel_launch(void* const* d_in, const int* in_sizes, int n_in,
                              void* d_out, int out_size, void* d_ws, size_t ws_size,
                              hipStream_t stream) {
    (void)in_sizes; (void)n_in; (void)out_size; (void)ws_size;

    // setup_inputs order: imagex (complex128), imagey (complex128), V (float64)
    const cplx*   imx = (const cplx*)d_in[0];
    const cplx*   imy = (const cplx*)d_in[1];
    const double* V   = (const double*)d_in[2];

    double* tab = (double*)d_ws;                      // 784*6 doubles = 37,632 B

    // outputs concatenated flat in return order: outputx then outputy,
    // each BATCH*IMG*IMG complex128 (interleaved doubles).
    cplx* outx = (cplx*)d_out;
    cplx* outy = outx + (size_t)BATCH * IMG * IMG;

    lc_coef_kernel <<<(NCELL * NCELL + 255) / 256, 256, 0, stream>>>(V, tab);
    lc_apply_kernel<<<BATCH * IMG, 256, 0, stream>>>(imx, imy, tab, outx, outy);
}